// Mapping_25890062860344
// MI455X (gfx1250) — compile-verified
//
#include <hip/hip_runtime.h>

namespace {
constexpr int G  = 256;
constexpr int E  = 64;
constexpr int D  = 64;
constexpr int BS = 8;
constexpr float EPSV = -1e16f;
constexpr int NP = BS * E * E * D;       // 2,097,152 (proj / proj_rot / crop / final)
constexpr int NF = BS * G * G * D;       // 33,554,432 (fgm)
constexpr size_t OFF_ROT = (size_t)NP;   // float offset of proj_rot inside ws
}

// ---- monotone float<->uint encoding so atomicMax(u32) == float max ----
__device__ __forceinline__ unsigned f2o(float f) {
  unsigned b = __float_as_uint(f);
  return (b & 0x80000000u) ? ~b : (b | 0x80000000u);
}
__device__ __forceinline__ float o2f(unsigned o) {
  unsigned b = (o & 0x80000000u) ? (o ^ 0x80000000u) : ~o;
  return __uint_as_float(b);
}

__device__ __forceinline__ void get_shifts(const float* gps, int b, float& sx, float& sy) {
  float grid_x = rintf((12.8f - gps[2 * b + 0]) / 0.1f);
  float grid_y = rintf((gps[2 * b + 1] + 12.8f) / 0.1f);
  sx = 128.0f - grid_y;   // == 128*tx (exact integer)
  sy = 128.0f - grid_x;   // == 128*ty (exact integer)
}

// K0: init encoded proj grid to enc(EPS)
__global__ void k_init(unsigned* __restrict__ projE) {
  int i = blockIdx.x * 256 + threadIdx.x;
  if (i < NP) projE[i] = f2o(EPSV);
}

// K1: depth -> ground-plane scatter-max of features into proj (encoded)
__global__ void k_scatter(const float* __restrict__ features,
                          const float* __restrict__ depth,
                          unsigned* __restrict__ projE) {
  int tid = blockIdx.x * 256 + threadIdx.x;
  if (tid >= NP) return;
  int p = tid & 4095;            // point (h,w) in 64x64
  int c = (tid >> 12) & 63;
  int b = tid >> 18;
  int h = p >> 6, w = p & 63;
  int hd = 4 * h, wd = 4 * w;    // stride-4 subsample of 256x256 depth
  float Z = depth[((size_t)b * 256 + hd) * 256 + wd] * 10.0f;
  float yy = (128.0f - (float)hd) / 128.0f;
  float Y = yy * Z;
  bool valid = (Z != 0.0f) && (Y > -1.5f) && (Y < 0.1f);
  float xx = ((float)wd - 128.0f) / 128.0f;
  float X = xx * Z;
  int x_gp = (int)rintf(X / 0.1f + 31.5f);
  int y_gp = (int)rintf(-(Z / 0.1f) + 31.5f);
  if (valid && x_gp >= 0 && x_gp < E && y_gp >= 0 && y_gp < E) {
    unsigned enc = f2o(features[tid]);  // features layout (b,c,h,w) == tid order
    atomicMax(&projE[(((size_t)b * E + y_gp) * E + x_gp) * D + c], enc);
  }
}

// K2: rotate proj by -compass (bilinear), EPS->0 on read. Channel-last in/out.
__global__ void k_rotate_proj(const unsigned* __restrict__ projE,
                              const float* __restrict__ compass,
                              float* __restrict__ rot) {
  int tid = blockIdx.x * 256 + threadIdx.x;   // float4 index
  if (tid >= NP / 4) return;
  int c4 = tid & 15, w = (tid >> 4) & 63, h = (tid >> 10) & 63, b = tid >> 16;
  float cm = compass[b];
  float ct = cosf(cm), st = -sinf(cm);        // t = -compass
  float xs = (2.0f * w + 1.0f) / 64.0f - 1.0f;
  float ys = (2.0f * h + 1.0f) / 64.0f - 1.0f;
  float gx = ct * xs + st * ys;
  float gy = -st * xs + ct * ys;
  float x = (gx + 1.0f) * 32.0f - 0.5f;
  float y = (gy + 1.0f) * 32.0f - 0.5f;
  float fx = floorf(x), fy = floorf(y);
  float wx = x - fx, wy = y - fy;
  int x0 = (int)fx, y0 = (int)fy;
  const uint4* pe4 = (const uint4*)projE;
  auto tap = [&](int ix, int iy) -> float4 {
    float4 v = make_float4(0.f, 0.f, 0.f, 0.f);
    if (ix < 0 || ix >= E || iy < 0 || iy >= E) return v;
    uint4 u = pe4[(((size_t)b * E + iy) * E + ix) * 16 + c4];
    float a;
    a = o2f(u.x); v.x = (a == EPSV) ? 0.0f : a;
    a = o2f(u.y); v.y = (a == EPSV) ? 0.0f : a;
    a = o2f(u.z); v.z = (a == EPSV) ? 0.0f : a;
    a = o2f(u.w); v.w = (a == EPSV) ? 0.0f : a;
    return v;
  };
  float4 v00 = tap(x0, y0), v10 = tap(x0 + 1, y0);
  float4 v01 = tap(x0, y0 + 1), v11 = tap(x0 + 1, y0 + 1);
  float w00 = (1.f - wx) * (1.f - wy), w10 = wx * (1.f - wy);
  float w01 = (1.f - wx) * wy, w11 = wx * wy;
  float4 r;
  r.x = v00.x * w00 + v10.x * w10 + v01.x * w01 + v11.x * w11;
  r.y = v00.y * w00 + v10.y * w10 + v01.y * w01 + v11.y * w11;
  r.z = v00.z * w00 + v10.z * w10 + v01.z * w01 + v11.z * w11;
  r.w = v00.w * w00 + v10.w * w10 + v01.w * w01 + v11.w * w11;
  ((float4*)rot)[tid] = r;
}

// K3: translated agent view + max with masked global map -> fgm output.
// One block per (b, gy). Stages the two needed proj_rot rows in LDS via
// CDNA5 async global->LDS loads (ASYNCcnt path), GV addressing form.
__global__ void k_translate_max(const float* __restrict__ rot,
                                const float* __restrict__ fullmap,
                                const float* __restrict__ gps,
                                const float* __restrict__ masks,
                                float* __restrict__ fgmOut) {
  __shared__ float4 lds[2][E * 16];   // 2 rows * 64 px * 64 ch = 32 KB
  int b = blockIdx.x >> 8;
  int gy = blockIdx.x & 255;
  float sx, sy;
  get_shifts(gps, b, sx, sy);
  float mask = masks[b];

  float srcy = (float)gy + sy;
  float fy = floorf(srcy);
  float wy = srcy - fy;
  int iy0 = (int)fy;

  const float4* rot4 = (const float4*)rot;
  for (int r = 0; r < 2; ++r) {
    int ry = (iy0 + r) - 96;           // row inside 64x64 agent window
    if (ry >= 0 && ry < E) {
      const float4* rowp = rot4 + ((size_t)b * E + ry) * (E * 16);
#if defined(__HIP_DEVICE_COMPILE__)
      for (int k = 0; k < 4; ++k) {
        int idx = threadIdx.x + k * 256;
        unsigned ldsOff = (unsigned)(unsigned long long)(void*)&lds[r][idx];
        const float4* srcp = rowp + idx;   // full 64-bit address in VGPR pair
        asm volatile("global_load_async_to_lds_b128 %0, %1, off"
                     :: "v"(ldsOff), "v"(srcp) : "memory");
      }
#else
      for (int k = 0; k < 4; ++k) {
        int idx = threadIdx.x + k * 256;
        lds[r][idx] = rowp[idx];
      }
#endif
    } else {
      for (int k = 0; k < 4; ++k)
        lds[r][threadIdx.x + k * 256] = make_float4(0.f, 0.f, 0.f, 0.f);
    }
  }
#if defined(__HIP_DEVICE_COMPILE__)
  asm volatile("s_wait_asynccnt 0" ::: "memory");
#endif
  __syncthreads();

  const float4* gm4 = (const float4*)fullmap;
  float4* out4 = (float4*)fgmOut;
  for (int it = 0; it < 16; ++it) {
    int e = threadIdx.x + it * 256;    // 0..4095 = (gx, c4)
    int c4 = e & 15;
    int gx = e >> 4;
    float srcx = (float)gx + sx;
    float fx = floorf(srcx);
    float wx = srcx - fx;
    int ix0 = (int)fx;
    int ca = ix0 - 96, cb = ca + 1;
    float4 z = make_float4(0.f, 0.f, 0.f, 0.f);
    float4 t00 = (ca >= 0 && ca < E) ? lds[0][ca * 16 + c4] : z;
    float4 t10 = (cb >= 0 && cb < E) ? lds[0][cb * 16 + c4] : z;
    float4 t01 = (ca >= 0 && ca < E) ? lds[1][ca * 16 + c4] : z;
    float4 t11 = (cb >= 0 && cb < E) ? lds[1][cb * 16 + c4] : z;
    float w00 = (1.f - wx) * (1.f - wy), w10 = wx * (1.f - wy);
    float w01 = (1.f - wx) * wy, w11 = wx * wy;
    size_t oi = (((size_t)b * G + gy) * G + gx) * 16 + c4;
    float4 g = gm4[oi];
    float4 o;
    o.x = fmaxf(g.x * mask, t00.x * w00 + t10.x * w10 + t01.x * w01 + t11.x * w11);
    o.y = fmaxf(g.y * mask, t00.y * w00 + t10.y * w10 + t01.y * w01 + t11.y * w11);
    o.z = fmaxf(g.z * mask, t00.z * w00 + t10.z * w10 + t01.z * w01 + t11.z * w11);
    o.w = fmaxf(g.w * mask, t00.w * w00 + t10.w * w10 + t01.w * w01 + t11.w * w11);
    out4[oi] = o;
  }
}

// K4: inverse-translate fgm and crop central 64x64 window (channel-last).
__global__ void k_crop(const float* __restrict__ fgm,
                       const float* __restrict__ gps,
                       float* __restrict__ crop) {
  int tid = blockIdx.x * 256 + threadIdx.x;  // float4 index
  if (tid >= NP / 4) return;
  int c4 = tid & 15, w = (tid >> 4) & 63, h = (tid >> 10) & 63, b = tid >> 16;
  float sx, sy;
  get_shifts(gps, b, sx, sy);
  float srcx = (float)(w + 96) - sx;
  float srcy = (float)(h + 96) - sy;
  float fx = floorf(srcx), fy = floorf(srcy);
  float wx = srcx - fx, wy = srcy - fy;
  int x0 = (int)fx, y0 = (int)fy;
  const float4* f4 = (const float4*)fgm;
  auto tap = [&](int ix, int iy) -> float4 {
    if (ix < 0 || ix >= G || iy < 0 || iy >= G) return make_float4(0.f, 0.f, 0.f, 0.f);
    return f4[(((size_t)b * G + iy) * G + ix) * 16 + c4];
  };
  float4 v00 = tap(x0, y0), v10 = tap(x0 + 1, y0);
  float4 v01 = tap(x0, y0 + 1), v11 = tap(x0 + 1, y0 + 1);
  float w00 = (1.f - wx) * (1.f - wy), w10 = wx * (1.f - wy);
  float w01 = (1.f - wx) * wy, w11 = wx * wy;
  float4 r;
  r.x = v00.x * w00 + v10.x * w10 + v01.x * w01 + v11.x * w11;
  r.y = v00.y * w00 + v10.y * w10 + v01.y * w01 + v11.y * w11;
  r.z = v00.z * w00 + v10.z * w10 + v01.z * w01 + v11.z * w11;
  r.w = v00.w * w00 + v10.w * w10 + v01.w * w01 + v11.w * w11;
  ((float4*)crop)[tid] = r;
}

// K5: rotate crop by +compass, write final in (b,c,h,w) order.
__global__ void k_rotate_final(const float* __restrict__ crop,
                               const float* __restrict__ compass,
                               float* __restrict__ finalOut) {
  int tid = blockIdx.x * 256 + threadIdx.x;
  if (tid >= NP) return;
  int w = tid & 63, h = (tid >> 6) & 63, c = (tid >> 12) & 63, b = tid >> 18;
  float cm = compass[b];
  float ct = cosf(cm), st = sinf(cm);        // t = +compass
  float xs = (2.0f * w + 1.0f) / 64.0f - 1.0f;
  float ys = (2.0f * h + 1.0f) / 64.0f - 1.0f;
  float gx = ct * xs + st * ys;
  float gy = -st * xs + ct * ys;
  float x = (gx + 1.0f) * 32.0f - 0.5f;
  float y = (gy + 1.0f) * 32.0f - 0.5f;
  float fx = floorf(x), fy = floorf(y);
  float wx = x - fx, wy = y - fy;
  int x0 = (int)fx, y0 = (int)fy;
  auto tap = [&](int ix, int iy) -> float {
    if (ix < 0 || ix >= E || iy < 0 || iy >= E) return 0.0f;
    return crop[(((size_t)b * E + iy) * E + ix) * D + c];
  };
  float v00 = tap(x0, y0), v10 = tap(x0 + 1, y0);
  float v01 = tap(x0, y0 + 1), v11 = tap(x0 + 1, y0 + 1);
  float w00 = (1.f - wx) * (1.f - wy), w10 = wx * (1.f - wy);
  float w01 = (1.f - wx) * wy, w11 = wx * wy;
  finalOut[tid] = v00 * w00 + v10 * w10 + v01 * w01 + v11 * w11;
}

extern "C" void kernel_launch(void* const* d_in, const int* in_sizes, int n_in,
                              void* d_out, int out_size, void* d_ws, size_t ws_size,
                              hipStream_t stream) {
  (void)in_sizes; (void)n_in; (void)out_size; (void)ws_size;
  const float* features = (const float*)d_in[0];
  const float* fullmap  = (const float*)d_in[1];
  const float* gps      = (const float*)d_in[2];
  const float* depth    = (const float*)d_in[3];
  const float* compass  = (const float*)d_in[4];
  const float* masks    = (const float*)d_in[5];

  float* ws       = (float*)d_ws;
  unsigned* projE = (unsigned*)d_ws;          // [0, NP) encoded scatter grid
  float* rot      = ws + OFF_ROT;             // [NP, 2NP) rotated agent view
  float* crop     = ws;                       // reuses projE region (done by then)

  float* finalOut = (float*)d_out;            // NP floats, (b,c,h,w)
  float* fgmOut   = (float*)d_out + NP;       // NF floats, (b,gy,gx,c)

  dim3 blk(256);
  k_init<<<dim3((NP + 255) / 256), blk, 0, stream>>>(projE);
  k_scatter<<<dim3((NP + 255) / 256), blk, 0, stream>>>(features, depth, projE);
  k_rotate_proj<<<dim3(NP / 4 / 256), blk, 0, stream>>>(projE, compass, rot);
  k_translate_max<<<dim3(BS * G), blk, 0, stream>>>(rot, fullmap, gps, masks, fgmOut);
  k_crop<<<dim3(NP / 4 / 256), blk, 0, stream>>>(fgmOut, gps, crop);
  k_rotate_final<<<dim3((NP + 255) / 256), blk, 0, stream>>>(crop, compass, finalOut);
}